// GraphNet_89051851915944
// MI455X (gfx1250) — compile-verified
//
#include <hip/hip_runtime.h>
#include <math.h>

// ---------------------------------------------------------------------------
// GraphNet (4-block ST-GCN) for MI455X / gfx1250, fp32 via V_WMMA_F32_16X16X4_F32
// Layout convention for all activations: [B, C, N] row-major (N = 340 nodes).
// ---------------------------------------------------------------------------

typedef __attribute__((ext_vector_type(2))) float v2f;
typedef __attribute__((ext_vector_type(4))) float v4f;
typedef __attribute__((ext_vector_type(8))) float v8f;

#define NNODES  340
#define TFRAMES 85
#define BATCHSZ 256
#define MTILES  22          // ceil(340/16)

__device__ __forceinline__ v8f wmma4(v2f a, v2f b, v8f c) {
  // D = A(16x4 f32) * B(4x16 f32) + C(16x16 f32); exact fp32.
  return __builtin_amdgcn_wmma_f32_16x16x4_f32(
      /*neg_a=*/false, a, /*neg_b=*/false, b,
      /*c_mod=*/(short)0, c, /*reuse_a=*/false, /*reuse_b=*/false);
}

// ---------------------------------------------------------------------------
// Segmented WMMA GEMM.
//   Out[b, o, n] over a 16(node) x 16(channel) tile per wave.
//   K dimension is a concatenation of up to 3 segments; segment s reads
//   A_s[b, c, n + noff_s] (zero outside [0,340)) against stacked weights
//   Bw[kglob, o] (row-major, Cout contiguous).
// EPI: 0 -> +bias ; 1 -> *scale + shift (conv bias + BN folded) ; 2 -> relu(+bias)
// ---------------------------------------------------------------------------
template <int EPI>
__global__ __launch_bounds__(256) void gemm_wmma(
    const float* __restrict__ A0, const float* __restrict__ A1,
    const float* __restrict__ A2,
    int kc0, int kc1, int kc2,            // # of 32-wide K chunks per segment
    int noff0, int noff1, int noff2,      // node offsets per segment
    const float* __restrict__ Bw,         // [32*(kc0+kc1+kc2)][Cout]
    const float* __restrict__ P0,         // bias (EPI 0/2) or BN scale (EPI 1)
    const float* __restrict__ P1,         // BN shift (EPI 1)
    int Cout, float* __restrict__ Out) {
  __shared__ float As[32][16];            // shared A tile (all 8 waves)
  __shared__ float Bs[8][32][16];         // per-wave B tile

  const int lane = threadIdx.x & 31;
  const int wave = threadIdx.x >> 5;
  const int b    = blockIdx.x / MTILES;
  const int n0   = (blockIdx.x % MTILES) * 16;
  const int o0   = (blockIdx.y * 8 + wave) * 16;

  v8f acc = {0.f, 0.f, 0.f, 0.f, 0.f, 0.f, 0.f, 0.f};

  const float* aps[3] = {A0, A1, A2};
  const int    kcs[3] = {kc0, kc1, kc2};
  const int    nos[3] = {noff0, noff1, noff2};

  int kglob = 0;
  const int m_a = lane & 15;
  const int ks  = (lane >> 4) << 1;       // 0 or 2 (K split across lane halves)

#pragma unroll
  for (int seg = 0; seg < 3; ++seg) {
    const float* Ap = aps[seg];
    const int kc = kcs[seg];
    if (kc == 0) continue;
    const int Kseg = kc * 32;
    const int noff = nos[seg];
    // Uniform fast path when the whole 16-node stripe is in range.
    const bool interior = (n0 + noff >= 0) && (n0 + 15 + noff < NNODES);
    for (int ch = 0; ch < kc; ++ch, ++kglob) {
      // --- cooperative A tile load: 512 floats by 256 threads ---
      if (interior) {
        const int e = threadIdx.x * 2;        // 2 consecutive floats per thread
        const int k = e >> 4, m = e & 15;     // m even -> 8B aligned both sides
        const float* src =
            &Ap[((size_t)b * Kseg + (ch * 32 + k)) * NNODES + (n0 + m + noff)];
        *(v2f*)&As[k][m] = *(const v2f*)src;
      } else {
#pragma unroll
        for (int r = 0; r < 2; ++r) {
          int e = threadIdx.x + r * 256;
          int k = e >> 4, m = e & 15;
          int n = n0 + m + noff;
          float v = 0.f;
          if (n >= 0 && n < NNODES)
            v = Ap[((size_t)b * Kseg + (ch * 32 + k)) * NNODES + n];
          As[k][m] = v;
        }
      }
      // --- per-wave B tile load: 512 floats, 4x b128 per lane ---
      {
        const float* Bp = Bw + (size_t)(kglob * 32) * Cout + o0;
#pragma unroll
        for (int r = 0; r < 4; ++r) {
          int e = lane * 4 + r * 128;
          int k = e >> 4, j = e & 15;         // j multiple of 4 -> 16B aligned
          *(v4f*)&Bs[wave][k][j] = *(const v4f*)&Bp[(size_t)k * Cout + j];
        }
      }
      __syncthreads();
      // --- prefetch all fragments, then 8 back-to-back WMMAs ---
      v2f afr[8], bfr[8];
#pragma unroll
      for (int s = 0; s < 8; ++s) {
        const int k0 = s * 4 + ks;
        afr[s].x = As[k0][m_a];
        afr[s].y = As[k0 + 1][m_a];
        bfr[s].x = Bs[wave][k0][m_a];
        bfr[s].y = Bs[wave][k0 + 1][m_a];
      }
#pragma unroll
      for (int s = 0; s < 8; ++s) acc = wmma4(afr[s], bfr[s], acc);
      __syncthreads();
    }
  }

  // --- epilogue: lane owns column o = o0 + (lane&15); rows mb..mb+7 ---
  const int oc = o0 + (lane & 15);
  const int mb = (lane >> 4) << 3;
  const float p0 = P0 ? P0[oc] : 0.f;
  const float p1 = (EPI == 1) ? P1[oc] : 0.f;

  float vals[8];
#pragma unroll
  for (int r = 0; r < 8; ++r) {
    float v = acc[r];
    if (EPI == 0) v += p0;
    if (EPI == 1) v = v * p0 + p1;
    if (EPI == 2) { v += p0; v = v > 0.f ? v : 0.f; }
    vals[r] = v;
  }

  // --- stage D through LDS (reuse this wave's B slice) for coalesced b128 store ---
  float* Dst = &Bs[wave][0][0];           // used as [16][20] (20 = aligned pitch)
#pragma unroll
  for (int r = 0; r < 8; ++r)
    Dst[(lane & 15) * 20 + (mb + r)] = vals[r];
  __syncthreads();
#pragma unroll
  for (int s = 0; s < 2; ++s) {
    const int idx2 = lane + s * 32;       // 64 float4 stores per tile
    const int orow = idx2 >> 2;
    const int q    = idx2 & 3;
    const int n    = n0 + q * 4;
    if (n < NNODES)                        // NNODES % 4 == 0 -> whole float4 valid
      *(v4f*)&Out[((size_t)b * Cout + (o0 + orow)) * NNODES + n] =
          *(const v4f*)&Dst[orow * 20 + q * 4];
  }
}

// ---------------------------------------------------------------------------
// GCNConv aggregation: out[b,o,n] = b1[o] + sum_{s in N(n) U {n}} dinv[s]*dinv[n]*in[b,o,s]
// Graph is the fixed 4-joint star + temporal chain (build_graph4).
// ---------------------------------------------------------------------------
__device__ __forceinline__ float deg_of(int t, int j) {
  return (j == 0 ? 4.f : 2.f) + (t > 0 ? 1.f : 0.f) +
         (t < TFRAMES - 1 ? 1.f : 0.f);
}

__global__ void gcn_aggregate(const float* __restrict__ hw,
                              const float* __restrict__ bias, int Cout,
                              float* __restrict__ out) {
  const int row = blockIdx.x;             // row = b*Cout + o
  const int o = row % Cout;
  const float* in = hw + (size_t)row * NNODES;
  float* op = out + (size_t)row * NNODES;
  const float bv = bias[o];
  for (int n = threadIdx.x; n < NNODES; n += blockDim.x) {
    const int t = n >> 2, j = n & 3;
    const float dn = rsqrtf(deg_of(t, j));
    float s = dn * dn * in[n];                                  // self loop
    if (j == 0) {
      const float ds = rsqrtf(deg_of(t, 1));                    // j=1,2,3 same deg
      s += dn * ds * (in[n + 1] + in[n + 2] + in[n + 3]);
    } else {
      s += dn * rsqrtf(deg_of(t, 0)) * in[t * 4];
    }
    if (t > 0)           s += dn * rsqrtf(deg_of(t - 1, j)) * in[n - 4];
    if (t < TFRAMES - 1) s += dn * rsqrtf(deg_of(t + 1, j)) * in[n + 4];
    op[n] = s + bv;
  }
}

// GraphConv neighbor sum (no self, no norm): g[b,c,n] = sum_{s in N(n)} h[b,c,s]
__global__ void sum_aggregate(const float* __restrict__ h,
                              float* __restrict__ out) {
  const int row = blockIdx.x;
  const float* in = h + (size_t)row * NNODES;
  float* op = out + (size_t)row * NNODES;
  for (int n = threadIdx.x; n < NNODES; n += blockDim.x) {
    const int t = n >> 2, j = n & 3;
    float s = (j == 0) ? (in[n + 1] + in[n + 2] + in[n + 3]) : in[t * 4];
    if (t > 0) s += in[n - 4];
    if (t < TFRAMES - 1) s += in[n + 4];
    op[n] = s;
  }
}

// ---------------------------------------------------------------------------
// Weight prep kernels (tiny, run once per block on-stream).
// ---------------------------------------------------------------------------
// Fuse conv1x1 + GCNConv weight:  Wf[c][o] = sum_j g1w[o][j]*convw[j][c]
//                                 bf[o]    = sum_j g1w[o][j]*convb[j]
__global__ void prep_fuse_conv_gcn(const float* __restrict__ convw,
                                   const float* __restrict__ convb,
                                   const float* __restrict__ g1w, int Cin,
                                   int Cout, float* __restrict__ Wf,
                                   float* __restrict__ bf) {
  const int idx = blockIdx.x * blockDim.x + threadIdx.x;
  if (idx < Cin * Cout) {
    const int o = idx % Cout, c = idx / Cout;
    float s = 0.f;
    for (int j = 0; j < Cout; ++j) s += g1w[o * Cout + j] * convw[j * Cin + c];
    Wf[c * Cout + o] = s;
  }
  if (idx < Cout) {
    float s = 0.f;
    for (int j = 0; j < Cout; ++j) s += g1w[idx * Cout + j] * convb[j];
    bf[idx] = s;
  }
}

// Temporal conv weights stacked [3*Cout][Cout]; fold conv bias + BN into sc/sh.
__global__ void prep_temporal(const float* __restrict__ tw,
                              const float* __restrict__ tb,
                              const float* __restrict__ gamma,
                              const float* __restrict__ beta,
                              const float* __restrict__ mean,
                              const float* __restrict__ var, int Cout,
                              float* __restrict__ Wt, float* __restrict__ sc,
                              float* __restrict__ sh) {
  const int idx = blockIdx.x * blockDim.x + threadIdx.x;
  if (idx < 3 * Cout * Cout) {
    const int o = idx % Cout;
    const int i = (idx / Cout) % Cout;
    const int kk = idx / (Cout * Cout);
    Wt[idx] = tw[(o * Cout + i) * 3 + kk];   // tconv_w is [O, I, 3, 1]
  }
  if (idx < Cout) {
    const float s = gamma[idx] * rsqrtf(var[idx] + 1e-5f);
    sc[idx] = s;
    sh[idx] = (tb[idx] - mean[idx]) * s + beta[idx];
  }
}

// GraphConv stacked weights: rows [0..Co)=Wr^T, [Co..2Co)=Wn^T,
// [2Co..2Co+Kres)= res-conv^T (block 3) or identity (residual pass-through).
__global__ void prep_graph2(const float* __restrict__ wr,
                            const float* __restrict__ wn,
                            const float* __restrict__ g2b,
                            const float* __restrict__ resw,
                            const float* __restrict__ resb, int Cout, int Kres,
                            float* __restrict__ WG, float* __restrict__ b2) {
  const int idx = blockIdx.x * blockDim.x + threadIdx.x;
  if (idx < Cout * Cout) {
    const int o = idx % Cout, i = idx / Cout;
    WG[(size_t)i * Cout + o] = wr[o * Cout + i];
    WG[(size_t)(Cout + i) * Cout + o] = wn[o * Cout + i];
  }
  if (idx < Kres * Cout) {
    const int o = idx % Cout, c = idx / Cout;
    const float v = resw ? resw[o * Kres + c] : ((c == o) ? 1.f : 0.f);
    WG[(size_t)(2 * Cout + c) * Cout + o] = v;
  }
  if (idx < Cout) b2[idx] = g2b[idx] + (resb ? resb[idx] : 0.f);
}

// ---------------------------------------------------------------------------
extern "C" void kernel_launch(void* const* d_in, const int* in_sizes, int n_in,
                              void* d_out, int out_size, void* d_ws,
                              size_t ws_size, hipStream_t stream) {
  (void)in_sizes; (void)n_in; (void)out_size; (void)ws_size;

  const float* x = (const float*)d_in[0];
  // setup_inputs order: x, then per block (dict insertion order):
  //   conv_w, conv_b, gcn1_w, gcn1_b, gcn2_wr, gcn2_wn, gcn2_b,
  //   tconv_w, tconv_b, bn_gamma, bn_beta, bn_mean, bn_var, [res_w, res_b]
  // then edge_src, edge_dst (unused: graph is static, baked into kernels).
  const bool hasres[4] = {false, false, true, false};
  const float* prm[4][15];
  int idx = 1;
  for (int b = 0; b < 4; ++b) {
    const int cnt = hasres[b] ? 15 : 13;
    for (int k = 0; k < cnt; ++k) prm[b][k] = (const float*)d_in[idx++];
  }

  // Workspace layout (floats).
  const size_t ACT = (size_t)BATCHSZ * 512 * NNODES;
  float* ws = (float*)d_ws;
  float* P0buf = ws;                  // ping activation buffer
  float* S1 = P0buf + ACT;            // hw / h3
  float* S2 = S1 + ACT;               // h2
  float* S3 = S2 + ACT;               // neighbor-sum g
  float* Wf = S3 + ACT;               // [512*512]
  float* bf = Wf + 512 * 512;
  float* Wt = bf + 512;               // [3*512*512]
  float* sc = Wt + 3 * 512 * 512;
  float* sh = sc + 512;
  float* WG = sh + 512;               // [(2*512+512)*512]
  float* b2 = WG + 3 * 512 * 512;

  const int Cin[4] = {256, 256, 256, 512};
  const int Co[4]  = {256, 256, 512, 512};
  const float* bin = x;
  float* bout[4] = {P0buf, (float*)d_out, P0buf, (float*)d_out};

  for (int blk = 0; blk < 4; ++blk) {
    const float** P = prm[blk];
    const int ci = Cin[blk], co = Co[blk];
    const int kres = hasres[blk] ? ci : co;

    prep_fuse_conv_gcn<<<(ci * co + 255) / 256, 256, 0, stream>>>(
        P[0], P[1], P[2], ci, co, Wf, bf);
    prep_temporal<<<(3 * co * co + 255) / 256, 256, 0, stream>>>(
        P[7], P[8], P[9], P[10], P[11], P[12], co, Wt, sc, sh);
    prep_graph2<<<(co * co + 255) / 256, 256, 0, stream>>>(
        P[4], P[5], P[6], hasres[blk] ? P[13] : nullptr,
        hasres[blk] ? P[14] : nullptr, co, kres, WG, b2);

    const dim3 gg(BATCHSZ * MTILES, co / 128);
    // hw = x @ (W1*Wc)^T + W1*bc            -> S1
    gemm_wmma<0><<<gg, 256, 0, stream>>>(bin, nullptr, nullptr, ci / 32, 0, 0,
                                         0, 0, 0, Wf, bf, nullptr, co, S1);
    // h2 = D^-1/2 (A+I) D^-1/2 hw + b1      -> S2
    gcn_aggregate<<<BATCHSZ * co, 128, 0, stream>>>(S1, P[3], co, S2);
    // h3 = BN(tconv(h2))  (3 shifted segs)  -> S1
    gemm_wmma<1><<<gg, 256, 0, stream>>>(S2, S2, S2, co / 32, co / 32, co / 32,
                                         -4, 0, 4, Wt, sc, sh, co, S1);
    // g = neighbor-sum(h3)                  -> S3
    sum_aggregate<<<BATCHSZ * co, 128, 0, stream>>>(S1, S3);
    // out = relu(h3 Wr^T + g Wn^T + res + b) -> bout
    gemm_wmma<2><<<gg, 256, 0, stream>>>(S1, S3, bin, co / 32, co / 32,
                                         kres / 32, 0, 0, 0, WG, b2, nullptr,
                                         co, bout[blk]);
    bin = bout[blk];
  }
}